// GINNet_69784628625692
// MI455X (gfx1250) — compile-verified
//
#include <hip/hip_runtime.h>

// GIN forward pass for MI455X (gfx1250), fp32 end-to-end.
// GEMMs: V_WMMA_F32_16X16X4_F32, LDS-staged A tiles, fragment-packed B,
// fused BatchNorm statistics in producer epilogues (saves 3 tensor passes/layer).

#define Nn 200000
#define Ee 400000
#define Gg 8000
#define Aa 100
#define Hh 128
#define Ll 4

typedef __attribute__((ext_vector_type(2))) float v2f;
typedef __attribute__((ext_vector_type(8))) float v8f;

__device__ __forceinline__ void atomAddF(float* p, float v) {
    __hip_atomic_fetch_add(p, v, __ATOMIC_RELAXED, __HIP_MEMORY_SCOPE_AGENT);
}

// ---------------- embedding: h0 = X[N,100] @ Wemb[100,128] -----------------
__global__ void embed_k(const float* __restrict__ X, const float* __restrict__ W,
                        float* __restrict__ h, float* __restrict__ h_in) {
    __shared__ float sh[Aa];
    int n = blockIdx.x, j = threadIdx.x;
    if (j < Aa) sh[j] = X[(long)n * Aa + j];
    __syncthreads();
    float acc = 0.f;
#pragma unroll 4
    for (int k = 0; k < Aa; ++k) acc = fmaf(sh[k], W[k * Hh + j], acc);
    long o = (long)n * Hh + j;
    h[o] = acc;
    h_in[o] = acc;
}

// ------ pack W[128,128] row-major -> WMMA-B fragment-major layout -----------
// pk[m][ks][nt][lane][c] = W[(4*ks + 2*(lane>>4) + c)*128 + nt*16 + (lane&15)]
// m: 0..3 = W1 layers, 4..7 = W2 layers, 8 = W_ro.
__global__ void pack_w_k(const float* __restrict__ W1, const float* __restrict__ W2,
                         const float* __restrict__ Wro, float* __restrict__ wpk) {
    long t = (long)blockIdx.x * 256 + threadIdx.x;   // 9*8192 fragments-pairs
    if (t >= 9L * 8192) return;
    int m = (int)(t >> 13);
    int r = (int)(t & 8191);
    int ks = r >> 8, nt = (r >> 5) & 7, lane = r & 31;
    int hi = lane >> 4, lo = lane & 15;
    const float* Ws = (m < 4) ? (W1 + (long)m * Hh * Hh)
                    : (m < 8) ? (W2 + (long)(m - 4) * Hh * Hh)
                              : Wro;
    int krow = 4 * ks + 2 * hi;
    int col = nt * 16 + lo;
    float* d = wpk + (long)m * 16384 + ((ks * 8 + nt) * 64 + lane * 2);
    d[0] = Ws[(long)krow * Hh + col];
    d[1] = Ws[(long)(krow + 1) * Hh + col];
}

// ------------- WMMA GEMM: Y = X[N,128] @ W + bias, optional fused stats -----
// 4 waves/block, 64-row tile staged in LDS (contiguous 32 KB), each wave:
// 16 rows x 128 cols = 8 v8f accumulators, K=128 -> 32 wmma issues per acc set.
// If stats != nullptr, accumulate per-column sum/sumsq of Y into stats[0..255]
// (register -> ds_add_f32 LDS reduction -> 1 global f32 atomic per column/block).
__global__ void __launch_bounds__(128) gemm128_k(const float* __restrict__ X,
                                                 const float* __restrict__ Wpk,
                                                 const float* __restrict__ bias,
                                                 float* __restrict__ Y,
                                                 float* __restrict__ stats) {
    __shared__ float lx[64 * Hh];      // 32 KB A tile
    __shared__ float sstat[2 * Hh];    // block-level sum / sumsq
    int tid = threadIdx.x;
    int wave = tid >> 5, lane = tid & 31;
    int lo = lane & 15, hi = lane >> 4;

    // stage the contiguous 64x128 tile with coalesced float4 copies
    {
        const float4* s4 = (const float4*)(X + (long)blockIdx.x * 64 * Hh);
        float4* d4 = (float4*)lx;
#pragma unroll
        for (int q = 0; q < 16; ++q) d4[q * 128 + tid] = s4[q * 128 + tid];
    }
    if (stats) { sstat[tid] = 0.f; sstat[tid + 128] = 0.f; }
    __syncthreads();

    v8f acc[8];
#pragma unroll
    for (int t = 0; t < 8; ++t) acc[t] = (v8f){0.f, 0.f, 0.f, 0.f, 0.f, 0.f, 0.f, 0.f};

    const float* arow = lx + (wave * 16 + lo) * Hh;
    for (int ks = 0; ks < 32; ++ks) {
        v2f a = *(const v2f*)(arow + ks * 4 + hi * 2);          // ds_load_b64
        const float* bbase = Wpk + ks * 512 + lane * 2;         // coalesced b64
#pragma unroll
        for (int nt = 0; nt < 8; ++nt) {
            v2f b = *(const v2f*)(bbase + nt * 64);
            acc[nt] = __builtin_amdgcn_wmma_f32_16x16x4_f32(
                false, a, false, b, (short)0, acc[nt], false, false);
        }
    }

    long row_base = (long)blockIdx.x * 64 + wave * 16;
#pragma unroll
    for (int nt = 0; nt < 8; ++nt) {
        int n = nt * 16 + lo;
        float bv = bias ? bias[n] : 0.f;
        float s = 0.f, s2 = 0.f;
#pragma unroll
        for (int r = 0; r < 8; ++r) {
            int m = r + hi * 8;
            float v = acc[nt][r] + bv;
            Y[(row_base + m) * Hh + n] = v;
            s += v;
            s2 = fmaf(v, v, s2);
        }
        if (stats) {
            atomicAdd(&sstat[n], s);          // ds_add_f32
            atomicAdd(&sstat[Hh + n], s2);
        }
    }
    if (stats) {
        __syncthreads();
        atomAddF(&stats[tid], sstat[tid]);            // tid 0..127: column sums
        atomAddF(&stats[Hh + tid], sstat[Hh + tid]);  // column sumsq
    }
}

// ---------------- x = (1+eps[layer]) * h ------------------------------------
__global__ void init_x_k(const float* __restrict__ h, float* __restrict__ x,
                         const float* __restrict__ eps, int layer, long total) {
    long i = (long)blockIdx.x * blockDim.x + threadIdx.x;
    if (i < total) x[i] = (1.f + eps[layer]) * h[i];
}

// ---------------- edge scatter: x[dst] += h[src] ----------------------------
__global__ void scatter_k(const float* __restrict__ h, float* __restrict__ x,
                          const int* __restrict__ src, const int* __restrict__ dst) {
    long i = (long)blockIdx.x * blockDim.x + threadIdx.x;
    long e = i >> 7;
    int j = (int)(i & 127);
    if (e < Ee) atomAddF(&x[(long)dst[e] * Hh + j], h[(long)src[e] * Hh + j]);
}

// ---------------- zero helper -----------------------------------------------
__global__ void zero_k(float* __restrict__ p, long total) {
    long i = (long)blockIdx.x * blockDim.x + threadIdx.x;
    if (i < total) p[i] = 0.f;
}

// ---- out = relu(bn(in)) [+resid]  (element-wise, no stats production) ------
__global__ void bn_apply_k(const float* __restrict__ in, float* __restrict__ out,
                           const float* __restrict__ stats,
                           const float* __restrict__ gamma, const float* __restrict__ beta,
                           const float* __restrict__ resid, long total) {
    long i = (long)blockIdx.x * blockDim.x + threadIdx.x;
    if (i >= total) return;
    int j = (int)(i & 127);
    const float invN = 1.f / (float)Nn;
    float mean = stats[j] * invN;
    float var = stats[Hh + j] * invN - mean * mean;
    float isd = rsqrtf(var + 1e-5f);
    float v = fmaf(gamma[j], (in[i] - mean) * isd, beta[j]);
    v = fmaxf(v, 0.f);
    if (resid) v += resid[i];
    out[i] = v;
}

// ---- out = relu(bn2(in)) * snorm[row], fused column stats of out -----------
__global__ void __launch_bounds__(256) bn_apply_stats_k(
    const float* __restrict__ in, float* __restrict__ out,
    const float* __restrict__ statsIn,
    const float* __restrict__ gamma, const float* __restrict__ beta,
    const float* __restrict__ snorm, float* __restrict__ statsOut) {
    __shared__ float sh[512];
    int j = threadIdx.x & 127, half = threadIdx.x >> 7;
    const float invN = 1.f / (float)Nn;
    float mean = statsIn[j] * invN;
    float var = statsIn[Hh + j] * invN - mean * mean;
    float isd = rsqrtf(var + 1e-5f);
    float ga = gamma[j], be = beta[j];
    float s = 0.f, s2 = 0.f;
    for (long r = (long)blockIdx.x * 2 + half; r < Nn; r += (long)gridDim.x * 2) {
        long i = r * Hh + j;
        float v = fmaf(ga, (in[i] - mean) * isd, be);
        v = fmaxf(v, 0.f) * snorm[r];
        out[i] = v;
        s += v;
        s2 = fmaf(v, v, s2);
    }
    sh[threadIdx.x] = s;
    sh[256 + threadIdx.x] = s2;
    __syncthreads();
    if (half == 0) {
        atomAddF(&statsOut[j], s + sh[128 + j]);
        atomAddF(&statsOut[Hh + j], s2 + sh[256 + 128 + j]);
    }
}

// ---------------- per-graph sum + counts ------------------------------------
__global__ void scatter_g_k(const float* __restrict__ hro, const int* __restrict__ gid,
                            float* __restrict__ gsum, float* __restrict__ gcnt) {
    int n = blockIdx.x, j = threadIdx.x;
    int g = gid[n];
    atomAddF(&gsum[(long)g * Hh + j], hro[(long)n * Hh + j]);
    if (j == 0) atomAddF(&gcnt[g], 1.f);
}

// ---------------- score[g] = (gsum[g]/max(cnt,1)) . Wpred + bpred -----------
__global__ void predict_k(const float* __restrict__ gsum, const float* __restrict__ gcnt,
                          const float* __restrict__ Wp, const float* __restrict__ bp,
                          float* __restrict__ out) {
    __shared__ float red[Hh];
    int g = blockIdx.x, j = threadIdx.x;
    float c = fmaxf(gcnt[g], 1.f);
    red[j] = gsum[(long)g * Hh + j] * Wp[j] / c;
    __syncthreads();
    for (int s = 64; s > 0; s >>= 1) {
        if (j < s) red[j] += red[j + s];
        __syncthreads();
    }
    if (j == 0) out[g] = red[0] + bp[0];
}

extern "C" void kernel_launch(void* const* d_in, const int* in_sizes, int n_in,
                              void* d_out, int out_size, void* d_ws, size_t ws_size,
                              hipStream_t stream) {
    const float* h0    = (const float*)d_in[0];
    const int*   src   = (const int*)d_in[1];
    const int*   dst   = (const int*)d_in[2];
    const int*   gid   = (const int*)d_in[3];
    const float* snorm = (const float*)d_in[4];
    const float* Wemb  = (const float*)d_in[5];
    const float* eps   = (const float*)d_in[6];
    const float* W1    = (const float*)d_in[7];
    const float* b1    = (const float*)d_in[8];
    const float* g1    = (const float*)d_in[9];
    const float* be1   = (const float*)d_in[10];
    const float* W2    = (const float*)d_in[11];
    const float* b2    = (const float*)d_in[12];
    const float* g2    = (const float*)d_in[13];
    const float* be2   = (const float*)d_in[14];
    const float* g3    = (const float*)d_in[15];
    const float* be3   = (const float*)d_in[16];
    const float* Wro   = (const float*)d_in[17];
    const float* Wp    = (const float*)d_in[18];
    const float* bp    = (const float*)d_in[19];
    float* out = (float*)d_out;
    (void)in_sizes; (void)n_in; (void)out_size; (void)ws_size;

    const long NH = (long)Nn * Hh;
    float* ws    = (float*)d_ws;
    float* h_in  = ws;                     // [N,H]
    float* h     = h_in + NH;              // [N,H]
    float* x     = h + NH;                 // [N,H]
    float* y     = x + NH;                 // [N,H]
    float* stats = y + NH;                 // 3 x 256 (sum | sumsq per column)
    float* gsum  = stats + 768;            // [G,H]
    float* gcnt  = gsum + (long)Gg * Hh;   // [G]
    float* wpack = gcnt + Gg;              // 9 x 16384 packed weights

    float* statsA = stats;
    float* statsB = stats + 256;
    float* statsC = stats + 512;

    const unsigned nhBlocks = (unsigned)((NH + 255) / 256);
    const unsigned edBlocks = (unsigned)(((long)Ee * Hh + 255) / 256);
    const unsigned gemmBlocks = Nn / 64;   // N is a multiple of 64

    pack_w_k<<<(9 * 8192 + 255) / 256, 256, 0, stream>>>(W1, W2, Wro, wpack);
    embed_k<<<Nn, Hh, 0, stream>>>(h0, Wemb, h, h_in);

    for (int i = 0; i < Ll; ++i) {
        zero_k<<<3, 256, 0, stream>>>(stats, 768);
        init_x_k<<<nhBlocks, 256, 0, stream>>>(h, x, eps, i, NH);
        scatter_k<<<edBlocks, 256, 0, stream>>>(h, x, src, dst);

        gemm128_k<<<gemmBlocks, 128, 0, stream>>>(x, wpack + (long)i * 16384,
                                                  b1 + i * Hh, y, statsA);
        bn_apply_k<<<nhBlocks, 256, 0, stream>>>(y, y, statsA, g1 + i * Hh, be1 + i * Hh,
                                                 nullptr, NH);

        gemm128_k<<<gemmBlocks, 128, 0, stream>>>(y, wpack + (long)(4 + i) * 16384,
                                                  b2 + i * Hh, x, statsB);
        bn_apply_stats_k<<<1024, 256, 0, stream>>>(x, x, statsB, g2 + i * Hh, be2 + i * Hh,
                                                   snorm, statsC);
        bn_apply_k<<<nhBlocks, 256, 0, stream>>>(x, h, statsC, g3 + i * Hh, be3 + i * Hh,
                                                 h_in, NH);
    }

    gemm128_k<<<gemmBlocks, 128, 0, stream>>>(h, wpack + 8L * 16384, nullptr, y, nullptr);
    zero_k<<<(unsigned)(((long)Gg * Hh + Gg + 255) / 256), 256, 0, stream>>>(
        gsum, (long)Gg * Hh + Gg);  // gcnt is contiguous after gsum
    scatter_g_k<<<Nn, Hh, 0, stream>>>(y, gid, gsum, gcnt);
    predict_k<<<Gg, Hh, 0, stream>>>(gsum, gcnt, Wp, bp, out);
}